// BiMpmMatching_25331717112141
// MI455X (gfx1250) — compile-verified
//
#include <hip/hip_runtime.h>
#include <hip/hip_bf16.h>
#include <float.h>

// ---------------------------------------------------------------------------
// BiMPM matching for MI455X (gfx1250): f32 WMMA (16x16x4) for all GEMM-like
// contractions, LDS-staged B operands shared across perspectives, wave32
// shuffle reductions elsewhere.
// ---------------------------------------------------------------------------

#define Bc 8
#define Sc 256
#define Hc 100
#define Pc 20
#define Cc 105
#define EPSF 1e-8f

typedef float v2f __attribute__((ext_vector_type(2)));
typedef float v8f __attribute__((ext_vector_type(8)));
typedef unsigned char uchar;

// workspace layout (floats)
#define OFF_COS      ((size_t)0)                         // B*S*S
#define OFF_N1       (OFF_COS + (size_t)Bc*Sc*Sc)        // B*S
#define OFF_N2       (OFF_N1 + (size_t)Bc*Sc)
#define OFF_ROWSUM   (OFF_N2 + (size_t)Bc*Sc)
#define OFF_COLSUM   (OFF_ROWSUM + (size_t)Bc*Sc)
#define OFF_N1P      (OFF_COLSUM + (size_t)Bc*Sc)        // B*P*S
#define OFF_N2P      (OFF_N1P + (size_t)Bc*Pc*Sc)
#define OFF_W2MP     (OFF_N2P + (size_t)Bc*Pc*Sc)        // P*H
#define OFF_C1LAST   (OFF_W2MP + (size_t)Pc*Hc)          // B*H
#define OFF_C2LAST   (OFF_C1LAST + (size_t)Bc*Hc)
#define OFF_LEN1     (OFF_C2LAST + (size_t)Bc*Hc)        // B
#define OFF_LEN2     (OFF_LEN1 + (size_t)Bc)
#define OFF_AM2      (OFF_LEN2 + (size_t)Bc)             // att_mean_2 B*S*H
#define OFF_AM1      (OFF_AM2 + (size_t)Bc*Sc*Hc)
#define OFF_AX2      (OFF_AM1 + (size_t)Bc*Sc*Hc)        // att_max_2
#define OFF_AX1      (OFF_AX2 + (size_t)Bc*Sc*Hc)

__device__ __forceinline__ v8f wmma_f32(v2f a, v2f b, v8f c) {
  // D = A(16x4) * B(4x16) + C ; 8-arg form: neg_a, A, neg_b, B, c_mod, C, reuse_a, reuse_b
  return __builtin_amdgcn_wmma_f32_16x16x4_f32(false, a, false, b, (short)0, c,
                                               false, false);
}

__device__ __forceinline__ float red32_sum(float v) {
  #pragma unroll
  for (int m = 16; m >= 1; m >>= 1) v += __shfl_xor(v, m, 32);
  return v;
}
__device__ __forceinline__ float red16_sum(float v) {
  #pragma unroll
  for (int m = 8; m >= 1; m >>= 1) v += __shfl_xor(v, m, 16);
  return v;
}
__device__ __forceinline__ float red16_max(float v) {
  #pragma unroll
  for (int m = 8; m >= 1; m >>= 1) v = fmaxf(v, __shfl_xor(v, m, 16));
  return v;
}

// ---------------------------------------------------------------------------
// w_maxpool^2 table
__global__ void k_prep_w2(const float* __restrict__ w, float* __restrict__ ws) {
  int i = blockIdx.x * blockDim.x + threadIdx.x;
  if (i < Pc * Hc) { float x = w[i]; ws[OFF_W2MP + i] = x * x; }
}

// per-(b,s) norms and per-perspective weighted norms; one wave per (b,s,side)
__global__ void k_prep_vec(const float* __restrict__ c1, const float* __restrict__ c2,
                           const uchar* __restrict__ m1, const uchar* __restrict__ m2,
                           const float* __restrict__ wmp, float* __restrict__ ws) {
  int s = blockIdx.x, b = blockIdx.y, side = blockIdx.z, l = threadIdx.x;
  const float* ctx = side ? c2 : c1;
  const uchar* msk = side ? m2 : m1;
  float mf = msk[b * Sc + s] ? 1.f : 0.f;
  const float* v = ctx + (size_t)(b * Sc + s) * Hc;
  float vh[4];
  float nn = 0.f;
  #pragma unroll
  for (int i = 0; i < 4; i++) {
    int h = l + 32 * i;
    float x = (h < Hc) ? v[h] * mf : 0.f;
    vh[i] = x; nn += x * x;
  }
  nn = red32_sum(nn);
  if (l == 0) ws[(side ? OFF_N2 : OFF_N1) + b * Sc + s] = fmaxf(sqrtf(nn), EPSF);
  float* np = ws + (side ? OFF_N2P : OFF_N1P);
  for (int p = 0; p < Pc; p++) {
    float acc = 0.f;
    #pragma unroll
    for (int i = 0; i < 4; i++) {
      int h = l + 32 * i;
      if (h < Hc) { float wv = wmp[p * Hc + h] * vh[i]; acc += wv * wv; }
    }
    acc = red32_sum(acc);
    if (l == 0) np[(size_t)(b * Pc + p) * Sc + s] = sqrtf(acc);  // unclamped
  }
}

// sequence lengths + last valid tokens
__global__ void k_prep_last(const float* __restrict__ c1, const float* __restrict__ c2,
                            const uchar* __restrict__ m1, const uchar* __restrict__ m2,
                            float* __restrict__ ws) {
  __shared__ int cnt1, cnt2;
  int b = blockIdx.x, t = threadIdx.x;
  if (t == 0) { cnt1 = 0; cnt2 = 0; }
  __syncthreads();
  if (m1[b * Sc + t]) atomicAdd(&cnt1, 1);
  if (m2[b * Sc + t]) atomicAdd(&cnt2, 1);
  __syncthreads();
  int i1 = max(cnt1 - 1, 0), i2 = max(cnt2 - 1, 0);
  if (t == 0) { ws[OFF_LEN1 + b] = (float)cnt1; ws[OFF_LEN2 + b] = (float)cnt2; }
  if (t < Hc) {
    float f1 = m1[b * Sc + i1] ? 1.f : 0.f;
    float f2 = m2[b * Sc + i2] ? 1.f : 0.f;
    ws[OFF_C1LAST + b * Hc + t] = c1[(size_t)(b * Sc + i1) * Hc + t] * f1;
    ws[OFF_C2LAST + b * Hc + t] = c2[(size_t)(b * Sc + i2) * Hc + t] * f2;
  }
}

// ---------------------------------------------------------------------------
// cos[b,s,t] = <c1_s, c2_t> / (n1*n2) via V_WMMA_F32_16X16X4_F32 (K = 25 * 4)
// grid(64, B), block 128 = 4 independent waves, one 16x16 tile each.
__global__ void k_cos_wmma(const float* __restrict__ c1, const float* __restrict__ c2,
                           const uchar* __restrict__ m1, const uchar* __restrict__ m2,
                           float* __restrict__ ws) {
  int wave = threadIdx.x >> 5, l = threadIdx.x & 31;
  int tile = blockIdx.x * 4 + wave;
  int b = blockIdx.y;
  int s0 = (tile >> 4) * 16, t0 = (tile & 15) * 16;
  int ls = l & 15, hg = l >> 4;
  float ma = m1[b * Sc + s0 + ls] ? 1.f : 0.f;
  float mb = m2[b * Sc + t0 + ls] ? 1.f : 0.f;
  const float* Ar = c1 + (size_t)(b * Sc + s0 + ls) * Hc;
  const float* Br = c2 + (size_t)(b * Sc + t0 + ls) * Hc;
  v8f acc = {};
  #pragma unroll
  for (int k = 0; k < 25; k++) {
    int ha = 4 * k + 2 * hg;
    v2f av; av.x = Ar[ha] * ma; av.y = Ar[ha + 1] * ma;
    v2f bv; bv.x = Br[4 * k + hg] * mb; bv.y = Br[4 * k + 2 + hg] * mb;
    acc = wmma_f32(av, bv, acc);
  }
  float n2v = ws[OFF_N2 + b * Sc + t0 + ls];
  float* cosm = ws + OFF_COS + (size_t)b * Sc * Sc;
  #pragma unroll
  for (int r = 0; r < 8; r++) {
    int s = s0 + r + 8 * hg;
    float n1v = ws[OFF_N1 + b * Sc + s];
    cosm[(size_t)s * Sc + t0 + ls] = acc[r] / (n1v * n2v);
  }
}

// masked max/mean + raw sums of cos rows (side 0 -> out1) / cols (side 1 -> out2)
__global__ void k_rowstats(const uchar* __restrict__ m1, const uchar* __restrict__ m2,
                           float* __restrict__ ws, float* __restrict__ out) {
  __shared__ float smax[256], ssum[256], sraw[256];
  int i = blockIdx.x, b = blockIdx.y, side = blockIdx.z, t = threadIdx.x;
  const float* cosm = ws + OFF_COS + (size_t)b * Sc * Sc;
  float v = side ? cosm[(size_t)t * Sc + i] : cosm[(size_t)i * Sc + t];
  const uchar* mo = side ? m1 : m2;
  bool mk = mo[b * Sc + t] != 0;
  smax[t] = mk ? v : -FLT_MAX;
  ssum[t] = mk ? v : 0.f;
  sraw[t] = v;
  __syncthreads();
  for (int off = 128; off > 0; off >>= 1) {
    if (t < off) {
      smax[t] = fmaxf(smax[t], smax[t + off]);
      ssum[t] += ssum[t + off];
      sraw[t] += sraw[t + off];
    }
    __syncthreads();
  }
  if (t == 0) {
    float cnt = fmaxf(ws[(side ? OFF_LEN1 : OFF_LEN2) + b], 1e-13f);
    float* o = out + (size_t)side * Bc * Sc * Cc + (size_t)(b * Sc + i) * Cc;
    o[0] = smax[0];
    o[1] = ssum[0] / cnt;
    ws[(side ? OFF_COLSUM : OFF_ROWSUM) + b * Sc + i] = sraw[0];
  }
}

// ---------------------------------------------------------------------------
// pairwise multi-perspective: one workgroup = 4 waves = 4 perspectives of the
// same (b, s-tile). Raw A tile and per-t-tile B tiles are staged in LDS once
// and shared by all 4 waves (w^2 is folded into A per wave). g via WMMA;
// masked max/mean over t reduced with 16-lane shuffles. Launched twice with
// roles swapped to cover out1 and out2.
__global__ void k_mv_wmma(const float* __restrict__ cA, const float* __restrict__ cB,
                          const uchar* __restrict__ mA, const uchar* __restrict__ mB,
                          const float* __restrict__ npA, const float* __restrict__ npB,
                          const float* __restrict__ lenB, const float* __restrict__ w2,
                          float* __restrict__ outSide) {
  __shared__ float sA[16 * Hc];
  __shared__ float sB[16 * Hc];
  int st = blockIdx.x, b = blockIdx.y;
  int tid = threadIdx.x;
  int wave = tid >> 5, l = tid & 31;
  int p = blockIdx.z * 4 + wave;
  int s0 = st * 16, ls = l & 15, hg = l >> 4;
  const float* Ab = cA + (size_t)b * Sc * Hc;
  const float* Bb = cB + (size_t)b * Sc * Hc;

  // stage masked A tile (p-invariant) once
  for (int idx = tid; idx < 16 * Hc; idx += 128) {
    int row = idx / Hc, h = idx - row * Hc;
    float mf = mA[b * Sc + s0 + row] ? 1.f : 0.f;
    sA[idx] = Ab[(size_t)(s0 + row) * Hc + h] * mf;
  }
  __syncthreads();

  // per-wave A operand with w^2 folded in
  v2f aw[25];
  #pragma unroll
  for (int k = 0; k < 25; k++) {
    int h = 4 * k + 2 * hg;
    aw[k].x = sA[ls * Hc + h] * w2[p * Hc + h];
    aw[k].y = sA[ls * Hc + h + 1] * w2[p * Hc + h + 1];
  }
  float n1v[8], runmax[8], runsum[8];
  #pragma unroll
  for (int r = 0; r < 8; r++) {
    n1v[r] = npA[(size_t)(b * Pc + p) * Sc + s0 + r + 8 * hg];
    runmax[r] = -FLT_MAX; runsum[r] = 0.f;
  }

  for (int tt = 0; tt < 16; tt++) {
    int t0 = tt * 16;
    __syncthreads();          // all waves done reading previous sB
    for (int idx = tid; idx < 16 * Hc; idx += 128) {
      int row = idx / Hc, h = idx - row * Hc;
      float mf = mB[b * Sc + t0 + row] ? 1.f : 0.f;
      sB[idx] = Bb[(size_t)(t0 + row) * Hc + h] * mf;
    }
    if (tt < 15 && tid < 64) {  // prefetch next tile (global_prefetch_b8)
      int row = tid >> 2, seg = tid & 3;
      __builtin_prefetch(Bb + (size_t)(t0 + 16 + row) * Hc + seg * 32, 0, 0);
    }
    __syncthreads();

    float mb = mB[b * Sc + t0 + ls] ? 1.f : 0.f;
    float n2v = npB[(size_t)(b * Pc + p) * Sc + t0 + ls];
    v8f acc = {};
    #pragma unroll
    for (int k = 0; k < 25; k++) {
      v2f bv;
      bv.x = sB[ls * Hc + 4 * k + hg];
      bv.y = sB[ls * Hc + 4 * k + 2 + hg];
      acc = wmma_f32(aw[k], bv, acc);
    }
    #pragma unroll
    for (int r = 0; r < 8; r++) {
      float mv = acc[r] / fmaxf(n1v[r] * n2v, EPSF);
      runmax[r] = fmaxf(runmax[r], mb > 0.f ? mv : -FLT_MAX);
      runsum[r] += mb * mv;
    }
  }
  float cnt = fmaxf(lenB[b], 1e-13f);
  #pragma unroll
  for (int r = 0; r < 8; r++) {
    float mx = red16_max(runmax[r]);
    float sm = red16_sum(runsum[r]);
    if (ls == 0) {
      int s = s0 + r + 8 * hg;
      float* o = outSide + (size_t)(b * Sc + s) * Cc;
      o[23 + p] = mx;
      o[43 + p] = sm / cnt;
    }
  }
}

// ---------------------------------------------------------------------------
// attentive mean vectors: att_mean_2 = (cos . c2) / rowsum ; att_mean_1 uses
// cos^T and c1 / colsum. M=256, K=256, N=100 (7 n-tiles, last partial).
__global__ void k_attmean_wmma(const float* __restrict__ c1, const float* __restrict__ c2,
                               float* __restrict__ ws) {
  int mt = blockIdx.x / 7, nt = blockIdx.x % 7;
  int b = blockIdx.y, side = blockIdx.z, l = threadIdx.x;
  int m0 = mt * 16, n0 = nt * 16, ls = l & 15, hg = l >> 4;
  const float* cosm = ws + OFF_COS + (size_t)b * Sc * Sc;
  const float* V = (side ? c1 : c2) + (size_t)b * Sc * Hc;
  int h = n0 + ls;
  v8f acc = {};
  for (int k = 0; k < 64; k++) {
    v2f av, bv;
    if (side == 0) {
      const float* cr = cosm + (size_t)(m0 + ls) * Sc + 4 * k + 2 * hg;
      av.x = cr[0]; av.y = cr[1];
    } else {
      av.x = cosm[(size_t)(4 * k + 2 * hg) * Sc + m0 + ls];
      av.y = cosm[(size_t)(4 * k + 2 * hg + 1) * Sc + m0 + ls];
    }
    float bx = 0.f, by = 0.f;
    if (h < Hc) {
      bx = V[(size_t)(4 * k + hg) * Hc + h];
      by = V[(size_t)(4 * k + 2 + hg) * Hc + h];
    }
    bv.x = bx; bv.y = by;
    acc = wmma_f32(av, bv, acc);
  }
  float* dst = ws + (side ? OFF_AM1 : OFF_AM2) + (size_t)b * Sc * Hc;
  const float* den = ws + (side ? OFF_COLSUM : OFF_ROWSUM) + b * Sc;
  #pragma unroll
  for (int r = 0; r < 8; r++) {
    int m = m0 + r + 8 * hg;
    float d = den[m];
    d = (d > EPSF) ? d : EPSF;
    if (h < Hc) dst[(size_t)m * Hc + h] = acc[r] / d;
  }
}

// attentive max vectors (elementwise max over opposite sequence)
__global__ void k_attmax(const float* __restrict__ c1, const float* __restrict__ c2,
                         const uchar* __restrict__ m1, const uchar* __restrict__ m2,
                         float* __restrict__ ws) {
  int i = blockIdx.x, b = blockIdx.y, side = blockIdx.z, h = threadIdx.x;
  if (h >= Hc) return;
  const float* cosm = ws + OFF_COS + (size_t)b * Sc * Sc;
  const float* V = (side ? c1 : c2) + (size_t)b * Sc * Hc;
  const uchar* mo = side ? m1 : m2;
  float best = -FLT_MAX;
  for (int t = 0; t < Sc; t++) {
    if (!mo[b * Sc + t]) continue;
    float cv = side ? cosm[(size_t)t * Sc + i] : cosm[(size_t)i * Sc + t];
    best = fmaxf(best, cv * V[(size_t)t * Hc + h]);
  }
  float* dst = ws + (side ? OFF_AX1 : OFF_AX2);
  dst[(size_t)(b * Sc + i) * Hc + h] = best;
}

// ---------------------------------------------------------------------------
// full/att/maxatt matching: sim_single + P weighted cosines. One wave per
// (b, s, side, variant).
__global__ void k_match(const float* __restrict__ c1, const float* __restrict__ c2,
                        const uchar* __restrict__ m1, const uchar* __restrict__ m2,
                        const float* __restrict__ w_full, const float* __restrict__ w_att,
                        const float* __restrict__ w_maxatt,
                        float* __restrict__ ws, float* __restrict__ out) {
  int s = blockIdx.x, b = blockIdx.y;
  int side = blockIdx.z / 3, var = blockIdx.z % 3, l = threadIdx.x;
  const float* ctx = side ? c2 : c1;
  const uchar* mm = side ? m2 : m1;
  float mf = mm[b * Sc + s] ? 1.f : 0.f;
  const float* v1 = ctx + (size_t)(b * Sc + s) * Hc;
  const float* v2; const float* w; int chS, chW;
  if (var == 0)      { v2 = ws + (side ? OFF_C1LAST : OFF_C2LAST) + b * Hc;
                       w = w_full;   chS = 2;  chW = 3;  }
  else if (var == 1) { v2 = ws + (side ? OFF_AM1 : OFF_AM2) + (size_t)(b * Sc + s) * Hc;
                       w = w_att;    chS = 63; chW = 64; }
  else               { v2 = ws + (side ? OFF_AX1 : OFF_AX2) + (size_t)(b * Sc + s) * Hc;
                       w = w_maxatt; chS = 84; chW = 85; }
  float a[4], c[4];
  float dot = 0.f, na = 0.f, nb = 0.f;
  #pragma unroll
  for (int i = 0; i < 4; i++) {
    int h = l + 32 * i;
    float x = (h < Hc) ? v1[h] * mf : 0.f;
    float y = (h < Hc) ? v2[h] : 0.f;
    a[i] = x; c[i] = y;
    dot += x * y; na += x * x; nb += y * y;
  }
  dot = red32_sum(dot); na = red32_sum(na); nb = red32_sum(nb);
  float* o = out + (size_t)side * Bc * Sc * Cc + (size_t)(b * Sc + s) * Cc;
  if (l == 0) o[chS] = dot / (fmaxf(sqrtf(na), EPSF) * fmaxf(sqrtf(nb), EPSF));
  for (int p = 0; p < Pc; p++) {
    float wd = 0.f, w1 = 0.f, w2s = 0.f;
    #pragma unroll
    for (int i = 0; i < 4; i++) {
      int h = l + 32 * i;
      if (h < Hc) {
        float wv = w[p * Hc + h];
        float wx = wv * a[i], wy = wv * c[i];
        wd += wx * wy; w1 += wx * wx; w2s += wy * wy;
      }
    }
    wd = red32_sum(wd); w1 = red32_sum(w1); w2s = red32_sum(w2s);
    if (l == 0) o[chW + p] = wd / (fmaxf(sqrtf(w1), EPSF) * fmaxf(sqrtf(w2s), EPSF));
  }
}

// ---------------------------------------------------------------------------
extern "C" void kernel_launch(void* const* d_in, const int* in_sizes, int n_in,
                              void* d_out, int out_size, void* d_ws, size_t ws_size,
                              hipStream_t stream) {
  const float* c1       = (const float*)d_in[0];
  const uchar* m1       = (const uchar*)d_in[1];
  const float* c2       = (const float*)d_in[2];
  const uchar* m2       = (const uchar*)d_in[3];
  const float* w_full   = (const float*)d_in[4];
  const float* w_maxpool= (const float*)d_in[5];
  const float* w_att    = (const float*)d_in[6];
  const float* w_maxatt = (const float*)d_in[7];
  float* out = (float*)d_out;
  float* ws  = (float*)d_ws;

  k_prep_w2  <<<(Pc * Hc + 255) / 256, 256, 0, stream>>>(w_maxpool, ws);
  k_prep_vec <<<dim3(Sc, Bc, 2), 32, 0, stream>>>(c1, c2, m1, m2, w_maxpool, ws);
  k_prep_last<<<Bc, 256, 0, stream>>>(c1, c2, m1, m2, ws);
  k_cos_wmma <<<dim3(64, Bc), 128, 0, stream>>>(c1, c2, m1, m2, ws);
  k_rowstats <<<dim3(Sc, Bc, 2), 256, 0, stream>>>(m1, m2, ws, out);
  // out1: A = c1 (s), B = c2 (t); 4 perspectives per workgroup share LDS tiles
  k_mv_wmma<<<dim3(16, Bc, Pc / 4), 128, 0, stream>>>(c1, c2, m1, m2,
      ws + OFF_N1P, ws + OFF_N2P, ws + OFF_LEN2, ws + OFF_W2MP, out);
  // out2: roles swapped
  k_mv_wmma<<<dim3(16, Bc, Pc / 4), 128, 0, stream>>>(c2, c1, m2, m1,
      ws + OFF_N2P, ws + OFF_N1P, ws + OFF_LEN1, ws + OFF_W2MP,
      out + (size_t)Bc * Sc * Cc);
  k_attmean_wmma<<<dim3(112, Bc, 2), 32, 0, stream>>>(c1, c2, ws);
  k_attmax  <<<dim3(Sc, Bc, 2), 128, 0, stream>>>(c1, c2, m1, m2, ws);
  k_match   <<<dim3(Sc, Bc, 6), 32, 0, stream>>>(c1, c2, m1, m2,
      w_full, w_att, w_maxatt, ws, out);
}